// MessagePassing_29575144800265
// MI455X (gfx1250) — compile-verified
//
#include <hip/hip_runtime.h>

// MessagePassing: out = segment_sum(x[src], dst) over E edges, d=64 f32.
// L2-resident working set on MI455X (77MB << 192MB L2); bound by L2 gather
// bandwidth + f32 atomic throughput. Strategy:
//   - one edge per wave32: lane l handles feature pair 2l (float2 load, 2 atomics)
//   - edge_index (int64 [2,E]) staged into LDS via CDNA5 async DMA
//     (global_load_async_to_lds_b64), double-buffered with s_wait_asynccnt
//   - native non-returning GLOBAL_ATOMIC_ADD_F32 via unsafeAtomicAdd
//     (avoids CAS-loop expansion of plain atomicAdd)

#define TILE    256   // edges staged per block per iteration
#define THREADS 256   // 8 waves (wave32)
#define NWAVES  (THREADS / 32)

__device__ __forceinline__ void atomic_add_f32_hw(float* p, float v) {
  // Lowers to global_atomic_add_f32 (no return) on gfx1250.
  unsafeAtomicAdd(p, v);
}

__global__ void mp_zero_kernel(float4* __restrict__ out, int n4) {
  int i = blockIdx.x * blockDim.x + threadIdx.x;
  const int stride = gridDim.x * blockDim.x;
  const float4 z = make_float4(0.f, 0.f, 0.f, 0.f);
  for (; i < n4; i += stride) out[i] = z;
}

__global__ void __launch_bounds__(THREADS)
mp_edge_scatter_kernel(const float* __restrict__ x,
                       const long long* __restrict__ ei,   // [2, E] int64
                       float* __restrict__ out,
                       int E, int nTiles) {
  __shared__ long long s_src[2][TILE];
  __shared__ long long s_dst[2][TILE];

  const int tid  = threadIdx.x;
  const int wave = tid >> 5;
  const int lane = tid & 31;
  const int col  = lane * 2;          // feature pair handled by this lane

  const unsigned long long srcBase = (unsigned long long)(const void*)ei;
  const unsigned long long dstBase = srcBase + (unsigned long long)E * 8ull;

  // Async-DMA one tile of (src,dst) indices into LDS buffer B.
  // Each of the 256 lanes copies one edge's src and dst (8B each).
  // 2 async instructions per wave per tile -> ASYNCcnt bookkeeping below.
#define STAGE(T, B)                                                           \
  do {                                                                        \
    long long _e = (long long)(T) * TILE + tid;                               \
    if (_e >= E) _e = E - 1; /* clamp: tail slots staged but never consumed */\
    unsigned long long _ga = srcBase + (unsigned long long)_e * 8ull;         \
    unsigned long long _gb = dstBase + (unsigned long long)_e * 8ull;         \
    unsigned _la = (unsigned)(unsigned long long)&s_src[(B)][tid];            \
    unsigned _lb = (unsigned)(unsigned long long)&s_dst[(B)][tid];            \
    asm volatile("global_load_async_to_lds_b64 %0, %1, off"                   \
                 :: "v"(_la), "v"(_ga) : "memory");                           \
    asm volatile("global_load_async_to_lds_b64 %0, %1, off"                   \
                 :: "v"(_lb), "v"(_gb) : "memory");                           \
  } while (0)

  int t = blockIdx.x;
  if (t < nTiles) STAGE(t, 0);

  int buf = 0;
  for (; t < nTiles; t += gridDim.x) {
    const int tn = t + gridDim.x;
    if (tn < nTiles) {
      STAGE(tn, buf ^ 1);
      // 4 outstanding (2 for tile t, 2 for tile tn); async loads complete
      // in order, so ASYNCcnt <= 2 means tile t's LDS writes are done.
      asm volatile("s_wait_asynccnt 2" ::: "memory");
    } else {
      asm volatile("s_wait_asynccnt 0" ::: "memory");
    }
    __syncthreads();  // every wave waited its own ASYNCcnt -> tile visible

    const long long tileBase = (long long)t * TILE;
    // wave w consumes tile-local edges w, w+8, ... : 32 edges per wave.
    // Per edge: 256B row gathered as float2/lane (2x128B lines), then
    // 2 non-returning f32 atomics per lane into out[dst].
#pragma unroll 4
    for (int j = wave; j < TILE; j += NWAVES) {
      if (tileBase + j < E) {
        const int s = (int)s_src[buf][j];   // LDS broadcast read
        const int d = (int)s_dst[buf][j];
        const float2 v =
            *(const float2*)(x + (size_t)s * 64 + col);
        float* o = out + (size_t)d * 64 + col;
        atomic_add_f32_hw(o,     v.x);   // global_atomic_add_f32 (no return)
        atomic_add_f32_hw(o + 1, v.y);
      }
    }
    __syncthreads();  // all reads of `buf` done before it is re-staged
    buf ^= 1;
  }
#undef STAGE
}

extern "C" void kernel_launch(void* const* d_in, const int* in_sizes, int n_in,
                              void* d_out, int out_size, void* d_ws, size_t ws_size,
                              hipStream_t stream) {
  const float*     x   = (const float*)d_in[0];       // [N, 64] f32
  const long long* ei  = (const long long*)d_in[1];   // [2, E] int64 (reference dtype)
  float*           out = (float*)d_out;               // [N, 64] f32

  const int E  = in_sizes[1] / 2;
  const int n4 = out_size / 4;

  // Harness poisons d_out; we accumulate with atomics, so zero it first.
  mp_zero_kernel<<<2048, 256, 0, stream>>>((float4*)out, n4);

  const int nTiles = (E + TILE - 1) / TILE;   // 6250 tiles for E=1.6M
  int blocks = 1024;                          // ~6 tiles/block, double-buffered
  if (blocks > nTiles) blocks = nTiles;
  mp_edge_scatter_kernel<<<blocks, THREADS, 0, stream>>>(x, ei, out, E, nTiles);
}